// InstanceRouteOptimizationArea_74328704024697
// MI455X (gfx1250) — compile-verified
//
#include <hip/hip_runtime.h>
#include <hip/hip_bf16.h>

// ---------------- problem constants (match reference) ----------------
#define NUM_NODES   120000
#define NUM_MOVABLE 100000
#define NUM_NETS    100000
#define NUM_PINS    400000
#define NBX 256
#define NBY 256
#define XL 0.0f
#define YL 0.0f
#define BSX 3.90625f            // (1000-0)/256
#define BSY 3.90625f
#define UNIT_H_CAP 1.5f
#define UNIT_V_CAP 1.5f
#define MAX_RATE 2.0f
#define MIN_RATE 0.5f
#define EPSF 1e-06f

#define NETS_PAD 100096          // 100000 rounded up to multiple of 256 (and 64)
#define NSPLIT 25                // K-split factor for the WMMA GEMM
#define CHUNK 4032               // 63*64; last chunk = 3328 = 52*64 (all mult. of 64)

typedef __attribute__((ext_vector_type(2))) float v2f;
typedef __attribute__((ext_vector_type(8))) float v8f;

// ---------------- kernel 0: zero H and V accumulators ----------------
__global__ void zero_hv_kernel(float* __restrict__ H, float* __restrict__ V) {
    int i = blockIdx.x * blockDim.x + threadIdx.x;
    if (i < NBX * NBY) { H[i] = 0.0f; V[i] = 0.0f; }
}

// ---------------- kernel 1: per-net bbox + coefficients ----------------
// Pads nets [NUM_NETS, NETS_PAD) with coef=0 so the GEMM loop is guard-free.
__global__ void net_bbox_kernel(const float* __restrict__ pin_pos,
                                const int* __restrict__ netpin_start,
                                const int* __restrict__ flat_netpin,
                                float4* __restrict__ bbox,
                                float2* __restrict__ coef) {
    int n = blockIdx.x * blockDim.x + threadIdx.x;
    if (n >= NETS_PAD) return;
    float xmin = XL, xmax = XL, ymin = YL, ymax = YL, ch = 0.0f, cv = 0.0f;
    if (n < NUM_NETS) {
        int s = netpin_start[n];
        int e = netpin_start[n + 1];
        if (e > s) {
            xmin = 3.0e38f; xmax = -3.0e38f; ymin = 3.0e38f; ymax = -3.0e38f;
            for (int p = s; p < e; ++p) {
                int pin = flat_netpin[p];
                float px = pin_pos[pin];
                float py = pin_pos[NUM_PINS + pin];
                xmin = fminf(xmin, px); xmax = fmaxf(xmax, px);
                ymin = fminf(ymin, py); ymax = fmaxf(ymax, py);
            }
            float sx = xmax - xmin;
            float sy = ymax - ymin;
            ch = (sy > EPSF) ? (1.0f / sy) : 0.0f;   // 1/max(sy,EPS) with sy>EPS
            cv = (sx > EPSF) ? (1.0f / sx) : 0.0f;
        }
    }
    bbox[n] = make_float4(xmin, xmax, ymin, ymax);
    coef[n] = make_float2(ch, cv);
}

// ---------------- kernel 2: H/V via V_WMMA_F32_16X16X4_F32 ----------------
// Grid: (256 tiles [16x16 grid of 16x16 output tiles], NSPLIT k-chunks).
// Block: 256 threads = 8 waves. Each wave consumes 8 nets per iteration as
// two K=4 WMMA steps with INDEPENDENT accumulator pairs (breaks the C-matrix
// RAW chain so the matrix pipe can overlap). coef_h/coef_v are folded into B
// so one A fragment feeds both the H and V WMMAs. 8-wave LDS reduction, then
// f32 global atomics merge the NSPLIT partials.
__global__ void __launch_bounds__(256) hv_wmma_kernel(const float4* __restrict__ bbox,
                                                      const float2* __restrict__ coef,
                                                      float* __restrict__ H,
                                                      float* __restrict__ V) {
    __shared__ float sH[256];
    __shared__ float sV[256];
    int tid = threadIdx.x;
    sH[tid] = 0.0f;
    sV[tid] = 0.0f;
    __syncthreads();

    int lane = tid & 31;
    int wave = tid >> 5;
    int half = lane >> 4;      // 0: lanes 0-15, 1: lanes 16-31
    int l15  = lane & 15;

    int xt = (blockIdx.x >> 4) << 4;   // x-bin tile origin
    int yt = (blockIdx.x & 15) << 4;   // y-bin tile origin

    float ex  = XL + (float)(xt + l15) * BSX;  // lane's x-bin left edge (A)
    float exh = ex + BSX;
    float ey  = YL + (float)(yt + l15) * BSY;  // lane's y-bin left edge (B)
    float eyh = ey + BSY;

    int start = (int)blockIdx.y * CHUNK + wave * 8;
    int stop  = min((int)(blockIdx.y + 1) * CHUNK, NETS_PAD);

    v8f accH0 = {}, accH1 = {};
    v8f accV0 = {}, accV1 = {};

    for (int n8 = start; n8 < stop; n8 += 64) {   // 8 waves * 8 nets
        __builtin_prefetch(&bbox[n8 + 64], 0, 0);  // global_prefetch_b8 (speculative)

        // A layout (16x4 f32): lanes 0-15 -> K=0,1 ; lanes 16-31 -> K=2,3.
        // Assumed B layout (4x16 f32): VGPR0: lanes0-15 K=0, lanes16-31 K=2;
        // VGPR1: lanes0-15 K=1, lanes16-31 K=3 -> same net pair as A per lane.
#pragma unroll
        for (int u = 0; u < 2; ++u) {
            int n0 = n8 + u * 4 + half * 2;
            float4 b0 = bbox[n0];
            float4 b1 = bbox[n0 + 1];
            float2 c0 = coef[n0];
            float2 c1 = coef[n0 + 1];

            v2f A, Bh, Bv;
            float o;
            // A = ox(net, x-bin)
            o = fminf(b0.y, exh) - fmaxf(b0.x, ex); A.x = fmaxf(o, 0.0f);
            o = fminf(b1.y, exh) - fmaxf(b1.x, ex); A.y = fmaxf(o, 0.0f);
            // B = coef * oy(net, y-bin)
            o = fminf(b0.w, eyh) - fmaxf(b0.z, ey); o = fmaxf(o, 0.0f);
            Bh.x = o * c0.x; Bv.x = o * c0.y;
            o = fminf(b1.w, eyh) - fmaxf(b1.z, ey); o = fmaxf(o, 0.0f);
            Bh.y = o * c1.x; Bv.y = o * c1.y;

            if (u == 0) {
                accH0 = __builtin_amdgcn_wmma_f32_16x16x4_f32(false, A, false, Bh,
                                                              (short)0, accH0, false, false);
                accV0 = __builtin_amdgcn_wmma_f32_16x16x4_f32(false, A, false, Bv,
                                                              (short)0, accV0, false, false);
            } else {
                accH1 = __builtin_amdgcn_wmma_f32_16x16x4_f32(false, A, false, Bh,
                                                              (short)0, accH1, false, false);
                accV1 = __builtin_amdgcn_wmma_f32_16x16x4_f32(false, A, false, Bv,
                                                              (short)0, accV1, false, false);
            }
        }
    }

    v8f accH = accH0 + accH1;
    v8f accV = accV0 + accV1;

    // C/D layout: VGPR r -> M = r + 8*half, N = l15. Reduce 8 waves via LDS.
#pragma unroll
    for (int r = 0; r < 8; ++r) {
        int idx = ((r + 8 * half) << 4) | l15;
        atomicAdd(&sH[idx], accH[r]);
        atomicAdd(&sV[idx], accV[r]);
    }
    __syncthreads();

    int m  = tid >> 4;
    int nn = tid & 15;
    int g  = (xt + m) * NBY + (yt + nn);
    atomicAdd(&H[g], sH[tid]);
    atomicAdd(&V[g], sV[tid]);
}

// ---------------- kernel 3: utilization map ----------------
__global__ void util_kernel(const float* __restrict__ H,
                            const float* __restrict__ V,
                            float* __restrict__ util) {
    int i = blockIdx.x * blockDim.x + threadIdx.x;
    if (i >= NBX * NBY) return;
    const float inv_h = 1.0f / (BSX * BSY * UNIT_H_CAP);
    const float inv_v = 1.0f / (BSX * BSY * UNIT_V_CAP);
    float u = fmaxf(H[i] * inv_h, V[i] * inv_v);
    util[i] = fminf(fmaxf(u, MIN_RATE), MAX_RATE);
}

// ---------------- kernel 4: per-instance route area ----------------
__global__ void instance_kernel(const float* __restrict__ pos,
                                const float* __restrict__ node_size_x,
                                const float* __restrict__ node_size_y,
                                const float* __restrict__ util,
                                float* __restrict__ out) {
    int mi = blockIdx.x * blockDim.x + threadIdx.x;
    if (mi >= NUM_MOVABLE) return;
    float xl = pos[mi];
    float yl = pos[NUM_NODES + mi];
    float xh = xl + node_size_x[mi];
    float yh = yl + node_size_y[mi];

    int x0 = max(0, (int)floorf((xl - XL) / BSX));
    int x1 = min(NBX - 1, (int)floorf((xh - XL) / BSX));
    int y0 = max(0, (int)floorf((yl - YL) / BSY));
    int y1 = min(NBY - 1, (int)floorf((yh - YL) / BSY));

    float acc = 0.0f;
    for (int x = x0; x <= x1; ++x) {
        float exl = XL + (float)x * BSX;
        float ox = fmaxf(fminf(xh, exl + BSX) - fmaxf(xl, exl), 0.0f);
        if (ox <= 0.0f) continue;
        float row = 0.0f;
        for (int y = y0; y <= y1; ++y) {
            float eyl = YL + (float)y * BSY;
            float oy = fmaxf(fminf(yh, eyl + BSY) - fmaxf(yl, eyl), 0.0f);
            row += util[x * NBY + y] * oy;
        }
        acc += ox * row;
    }
    out[mi] = acc;
}

// ---------------- launch ----------------
extern "C" void kernel_launch(void* const* d_in, const int* in_sizes, int n_in,
                              void* d_out, int out_size, void* d_ws, size_t ws_size,
                              hipStream_t stream) {
    const float* pos          = (const float*)d_in[0];
    const float* pin_pos      = (const float*)d_in[1];
    const float* node_size_x  = (const float*)d_in[2];
    const float* node_size_y  = (const float*)d_in[3];
    const int*   netpin_start = (const int*)d_in[4];
    const int*   flat_netpin  = (const int*)d_in[5];
    float* out = (float*)d_out;

    // workspace layout (bytes), 16B aligned
    char* ws = (char*)d_ws;
    float4* bbox = (float4*)(ws);                              // NETS_PAD * 16B
    float2* coef = (float2*)(ws + (size_t)NETS_PAD * 16);      // NETS_PAD * 8B
    float*  H    = (float*)(ws + (size_t)NETS_PAD * 24);       // 256KB
    float*  V    = H + NBX * NBY;                              // 256KB
    float*  util = V + NBX * NBY;                              // 256KB

    zero_hv_kernel<<<(NBX * NBY + 255) / 256, 256, 0, stream>>>(H, V);

    net_bbox_kernel<<<NETS_PAD / 256, 256, 0, stream>>>(
        pin_pos, netpin_start, flat_netpin, bbox, coef);

    dim3 g2((NBX / 16) * (NBY / 16), NSPLIT);
    hv_wmma_kernel<<<g2, 256, 0, stream>>>(bbox, coef, H, V);

    util_kernel<<<(NBX * NBY + 255) / 256, 256, 0, stream>>>(H, V, util);

    instance_kernel<<<(NUM_MOVABLE + 255) / 256, 256, 0, stream>>>(
        pos, node_size_x, node_size_y, util, out);
}